// UncertaintyWeightedAttention_5617817223662
// MI455X (gfx1250) — compile-verified
//
#include <hip/hip_runtime.h>
#include <hip/hip_bf16.h>

typedef __attribute__((ext_vector_type(2))) float v2f;
typedef __attribute__((ext_vector_type(4))) float v4f;
typedef __attribute__((ext_vector_type(8))) float v8f;
typedef int v4i __attribute__((__vector_size__(16)));   // matches builtin's V4i

// Problem constants (from reference)
constexpr int CB  = 2;      // batch
constexpr int CS  = 2048;   // sequence
constexpr int CH  = 1024;   // hidden
constexpr int CNH = 16;     // heads
constexpr int CHD = 64;     // head dim
constexpr float SM_SCALE = 0.125f;     // 1/sqrt(64)
constexpr float LAMBDA   = 0.5f;

// ---------------------------------------------------------------------------
// Async global->LDS staging (CDNA5 GLOBAL_LOAD_ASYNC_TO_LDS_B128, ASYNCcnt),
// with a compile-safe fallback to load_b128 + ds_store_b128.
// ---------------------------------------------------------------------------
#define GLOBAL_AS __attribute__((address_space(1)))
#define LDS_AS    __attribute__((address_space(3)))

#if defined(__AMDGCN__) && \
    __has_builtin(__builtin_amdgcn_global_load_async_to_lds_b128) && \
    __has_builtin(__builtin_amdgcn_s_wait_asynccnt)
#define USE_ASYNC_LDS 1
#else
#define USE_ASYNC_LDS 0
#endif

__device__ __forceinline__ void cp16(float* lds_dst, const float* g_src) {
#if USE_ASYNC_LDS
  __builtin_amdgcn_global_load_async_to_lds_b128(
      (GLOBAL_AS v4i*)g_src, (LDS_AS v4i*)lds_dst, 0, 0);
#else
  *(v4f*)lds_dst = *(const v4f*)g_src;
#endif
}

__device__ __forceinline__ void async_join() {
#if USE_ASYNC_LDS
  __builtin_amdgcn_s_wait_asynccnt(0);
#endif
}

// ---------------------------------------------------------------------------
// Kernel 0: uncertainty weights w = exp(-lambda * U)
// ---------------------------------------------------------------------------
__global__ void wexp_kernel(const float* __restrict__ U, float* __restrict__ w, int n) {
  int i = blockIdx.x * blockDim.x + threadIdx.x;
  if (i < n) w[i] = __expf(-LAMBDA * U[i]);
}

// ---------------------------------------------------------------------------
// Kernel 1: C[M,N] = A[M,K] @ W[K,N] + bias[N]   (fp32 WMMA 16x16x4)
// Block: 256 threads = 8 waves (4 M x 2 N), block tile 128x128, BK=16,
// double-buffered LDS with async prefetch of the next K-stage.
// ---------------------------------------------------------------------------
constexpr int BM = 128, BN = 128, BK = 16;
constexpr int LDA = BK + 4;   // 20: b128-store aligned (80B), 8B frag reads, bank-spread
constexpr int LDB = BN + 4;   // 132: b128-store aligned (528B)

__global__ __launch_bounds__(256)
void gemm_bias_kernel(const float* __restrict__ A, const float* __restrict__ W,
                      const float* __restrict__ bias, float* __restrict__ C,
                      int M, int N, int K) {
  __shared__ float sA[2][BM * LDA];   // 2 x 10.0 KB
  __shared__ float sB[2][BK * LDB];   // 2 x  8.25 KB

  const int tid  = threadIdx.x;
  const int lane = tid & 31;
  const int wave = tid >> 5;
  const int wm   = wave & 3;      // 0..3 (M)
  const int wn   = wave >> 2;     // 0..1 (N)
  const int lo   = lane & 15;
  const int hi   = lane >> 4;     // half-wave select
  const int m_block = blockIdx.x * BM;
  const int n_block = blockIdx.y * BN;

  v8f acc[2][4] = {};               // wave tile 32(M) x 64(N)

  auto stage = [&](int kk, int bufi) {
    // A tile: 128 x 16 floats, 512 x 16B chunks, 2 per thread
    #pragma unroll
    for (int t = 0; t < 2; ++t) {
      int i   = tid + t * 256;
      int row = i >> 2;            // / (BK/4)
      int c4  = i & 3;
      cp16(&sA[bufi][row * LDA + c4 * 4],
           A + (size_t)(m_block + row) * K + kk + c4 * 4);
    }
    // B tile: 16 x 128 floats, 512 x 16B chunks, 2 per thread
    #pragma unroll
    for (int t = 0; t < 2; ++t) {
      int i   = tid + t * 256;
      int row = i >> 5;            // / (BN/4)
      int c4  = i & 31;
      cp16(&sB[bufi][row * LDB + c4 * 4],
           W + (size_t)(kk + row) * N + n_block + c4 * 4);
    }
  };

  stage(0, 0);
  async_join();
  __syncthreads();

  int cur = 0;
  for (int kk = 0; kk < K; kk += BK) {
    if (kk + BK < K) stage(kk + BK, cur ^ 1);   // prefetch next stage (async)

    #pragma unroll
    for (int k = 0; k < BK; k += 4) {
      v2f afrag[2], bfrag[4];
      #pragma unroll
      for (int i = 0; i < 2; ++i) {
        int m = wm * 32 + i * 16 + lo;
        afrag[i] = *(const v2f*)(&sA[cur][m * LDA + k + 2 * hi]);   // A[M=lo][K=k+2hi,+1]
      }
      #pragma unroll
      for (int j = 0; j < 4; ++j) {
        int n = wn * 64 + j * 16 + lo;
        float b0 = sB[cur][(k + 2 * hi)     * LDB + n];             // B[K=k+2hi  ][N=lo]
        float b1 = sB[cur][(k + 2 * hi + 1) * LDB + n];             // B[K=k+2hi+1][N=lo]
        bfrag[j] = (v2f){b0, b1};
      }
      #pragma unroll
      for (int i = 0; i < 2; ++i)
        #pragma unroll
        for (int j = 0; j < 4; ++j)
          acc[i][j] = __builtin_amdgcn_wmma_f32_16x16x4_f32(
              false, afrag[i], false, bfrag[j], (short)0, acc[i][j], false, false);
    }

    async_join();       // next-stage buffer fully landed in LDS
    __syncthreads();    // and everyone is done reading the current buffer
    cur ^= 1;
  }

  // Epilogue: C layout VGPR v -> M = v + 8*hi, N = lo
  #pragma unroll
  for (int i = 0; i < 2; ++i) {
    #pragma unroll
    for (int j = 0; j < 4; ++j) {
      int n = n_block + wn * 64 + j * 16 + lo;
      float bv = bias[n];
      #pragma unroll
      for (int vv = 0; vv < 8; ++vv) {
        int m = m_block + wm * 32 + i * 16 + vv + 8 * hi;
        C[(size_t)m * N + n] = acc[i][j][vv] + bv;
      }
    }
  }
}

// ---------------------------------------------------------------------------
// Kernel 2: flash-style uncertainty-weighted attention.
// Grid: (S/QT, NH, B). Block: 256 threads = 8 waves; each wave owns 16 q-rows.
// ---------------------------------------------------------------------------
constexpr int QT  = 128;         // query rows per block
constexpr int KT  = 32;          // key tile
constexpr int LKV = CHD + 4;     // 68
constexpr int LP  = KT + 4;      // 36

__global__ __launch_bounds__(256)
void attn_kernel(const float* __restrict__ q, const float* __restrict__ k,
                 const float* __restrict__ v, const float* __restrict__ wexp,
                 const int* __restrict__ amask, float* __restrict__ ctx) {
  __shared__ float sK[KT * LKV];   //  8.7 KB
  __shared__ float sV[KT * LKV];   //  8.7 KB
  __shared__ float sP[QT * LP];    // 18.4 KB

  const int tid  = threadIdx.x;
  const int lane = tid & 31;
  const int wave = tid >> 5;
  const int lo   = lane & 15;
  const int hi   = lane >> 4;
  const int q0   = blockIdx.x * QT;
  const int h    = blockIdx.y;
  const int b    = blockIdx.z;
  const size_t base = ((size_t)b * CS) * CH + (size_t)h * CHD;  // + s*CH + d

  // Preload this wave's 16 Q rows as A-fragments (16 k-steps over HD).
  v2f qf[CHD / 4];
  {
    const float* qrow = q + base + (size_t)(q0 + wave * 16 + lo) * CH;
    #pragma unroll
    for (int t = 0; t < CHD / 4; ++t)
      qf[t] = *(const v2f*)(qrow + 4 * t + 2 * hi);
  }

  v8f accO[4] = {};                  // 4 d-tiles of 16
  float mrow[8], lrow[8];
  #pragma unroll
  for (int vv = 0; vv < 8; ++vv) { mrow[vv] = -3.0e38f; lrow[vv] = 0.0f; }

  for (int kn0 = 0; kn0 < CS; kn0 += KT) {
    __syncthreads();                 // prior P@V reads done before restaging K/V
    // Stage K and V tiles: KT x HD floats each, 512 x 16B chunks per array
    #pragma unroll
    for (int t = 0; t < 2; ++t) {
      int i   = tid + t * 256;       // 0..511
      int row = i >> 4;              // / (HD/4)
      int c4  = i & 15;
      size_t g = base + (size_t)(kn0 + row) * CH + c4 * 4;
      cp16(&sK[row * LKV + c4 * 4], k + g);
      cp16(&sV[row * LKV + c4 * 4], v + g);
    }
    async_join();
    __syncthreads();

    // S = Q @ K^T  (2 n-tiles of 16 keys)
    v8f sc[2] = {};
    #pragma unroll
    for (int t = 0; t < CHD / 4; ++t) {
      #pragma unroll
      for (int j = 0; j < 2; ++j) {
        // B[d][n] = K[n][d]: lane needs K[j*16+lo][4t+2hi], K[j*16+lo][4t+2hi+1]
        v2f bf = *(const v2f*)(&sK[(j * 16 + lo) * LKV + 4 * t + 2 * hi]);
        sc[j] = __builtin_amdgcn_wmma_f32_16x16x4_f32(
            false, qf[t], false, bf, (short)0, sc[j], false, false);
      }
    }

    // scale * w_key, mask (each lane's 8 elements share key column N = lo)
    #pragma unroll
    for (int j = 0; j < 2; ++j) {
      int key  = kn0 + j * 16 + lo;
      float wj = SM_SCALE * wexp[b * CS + key];
      bool ok  = amask[b * CS + key] != 0;
      #pragma unroll
      for (int vv = 0; vv < 8; ++vv)
        sc[j][vv] = ok ? sc[j][vv] * wj : -3.0e38f;
    }

    // Online softmax: row r = vv + 8*hi lives across the 16 lanes of a half-wave
    float corr[8];
    #pragma unroll
    for (int vv = 0; vv < 8; ++vv) {
      float rm = fmaxf(sc[0][vv], sc[1][vv]);
      #pragma unroll
      for (int off = 8; off >= 1; off >>= 1)
        rm = fmaxf(rm, __shfl_xor(rm, off, 32));
      float mn = fmaxf(mrow[vv], rm);
      float p0 = __expf(sc[0][vv] - mn);
      float p1 = __expf(sc[1][vv] - mn);
      sc[0][vv] = p0; sc[1][vv] = p1;
      float rs = p0 + p1;
      #pragma unroll
      for (int off = 8; off >= 1; off >>= 1)
        rs += __shfl_xor(rs, off, 32);
      corr[vv] = __expf(mrow[vv] - mn);
      lrow[vv] = lrow[vv] * corr[vv] + rs;
      mrow[vv] = mn;
    }
    #pragma unroll
    for (int jd = 0; jd < 4; ++jd)
      #pragma unroll
      for (int vv = 0; vv < 8; ++vv)
        accO[jd][vv] *= corr[vv];

    // P: C-layout -> LDS -> A-fragment layout
    #pragma unroll
    for (int j = 0; j < 2; ++j)
      #pragma unroll
      for (int vv = 0; vv < 8; ++vv)
        sP[(wave * 16 + vv + 8 * hi) * LP + j * 16 + lo] = sc[j][vv];
    __syncthreads();

    // O += P @ V   (K-dim = KT keys, 4 d-tiles)
    #pragma unroll
    for (int kk = 0; kk < KT; kk += 4) {
      v2f af = *(const v2f*)(&sP[(wave * 16 + lo) * LP + kk + 2 * hi]);
      #pragma unroll
      for (int jd = 0; jd < 4; ++jd) {
        float b0 = sV[(kk + 2 * hi)     * LKV + jd * 16 + lo];  // V[key][d]
        float b1 = sV[(kk + 2 * hi + 1) * LKV + jd * 16 + lo];
        v2f bf = (v2f){b0, b1};
        accO[jd] = __builtin_amdgcn_wmma_f32_16x16x4_f32(
            false, af, false, bf, (short)0, accO[jd], false, false);
      }
    }
  }

  // Normalize and write ctx[b, s, h*HD + d]
  #pragma unroll
  for (int vv = 0; vv < 8; ++vv) {
    float inv = lrow[vv] > 0.0f ? 1.0f / lrow[vv] : 0.0f;
    int srow = q0 + wave * 16 + vv + 8 * hi;
    #pragma unroll
    for (int jd = 0; jd < 4; ++jd)
      ctx[base + (size_t)srow * CH + jd * 16 + lo] = accO[jd][vv] * inv;
  }
}

// ---------------------------------------------------------------------------
// Launcher
// ---------------------------------------------------------------------------
extern "C" void kernel_launch(void* const* d_in, const int* in_sizes, int n_in,
                              void* d_out, int out_size, void* d_ws, size_t ws_size,
                              hipStream_t stream) {
  (void)in_sizes; (void)n_in; (void)out_size; (void)ws_size;

  const float* X    = (const float*)d_in[0];   // hidden_states [B,S,H]
  const float* U    = (const float*)d_in[1];   // token_uncertainty [B,S]
  const int*   msk  = (const int*)  d_in[2];   // attention_mask [B,S]
  const float* Wq   = (const float*)d_in[3];
  const float* bq   = (const float*)d_in[4];
  const float* Wk   = (const float*)d_in[5];
  const float* bk   = (const float*)d_in[6];
  const float* Wv   = (const float*)d_in[7];
  const float* bv   = (const float*)d_in[8];
  const float* Wo   = (const float*)d_in[9];
  const float* bo   = (const float*)d_in[10];
  float* out = (float*)d_out;

  constexpr size_t BSH = (size_t)CB * CS * CH;   // 4,194,304 floats
  float* qb   = (float*)d_ws;
  float* kb   = qb + BSH;
  float* vb   = kb + BSH;
  float* ctx  = vb + BSH;
  float* wexp = ctx + BSH;                       // B*S floats

  const int M = CB * CS, N = CH, K = CH;
  dim3 gblk(256);
  dim3 ggrid(M / BM, N / BN);                    // 32 x 8

  wexp_kernel<<<(CB * CS + 255) / 256, 256, 0, stream>>>(U, wexp, CB * CS);

  gemm_bias_kernel<<<ggrid, gblk, 0, stream>>>(X, Wq, bq, qb, M, N, K);
  gemm_bias_kernel<<<ggrid, gblk, 0, stream>>>(X, Wk, bk, kb, M, N, K);
  gemm_bias_kernel<<<ggrid, gblk, 0, stream>>>(X, Wv, bv, vb, M, N, K);

  dim3 agrid(CS / QT, CNH, CB);                  // 16 x 16 x 2
  attn_kernel<<<agrid, gblk, 0, stream>>>(qb, kb, vb, wexp, msk, ctx);

  gemm_bias_kernel<<<ggrid, gblk, 0, stream>>>(ctx, Wo, bo, out, M, N, K);
}